// IDWT_2D_11501922418859
// MI455X (gfx1250) — compile-verified
//
#include <hip/hip_runtime.h>

// IDWT 2D (Haar, stride-2 transpose conv == pixel-shuffle scatter) for MI455X.
//
// Per 16 consecutive pixels this is one V_WMMA_F32_16X16X4_F32:
//   D[M=pixel, N=(j,a,c)] = sum_K A[M, K=group-channel] * B[K, N],  B[k,n] = filters[n]
// which fuses the 4-channel reduction (K=4) and the +-0.5 filter outer product
// (N=16) into a single matrix op. D is staged through LDS (bank-conflict-free,
// row stride 33 dwords) to turn the scattered (j,a,c) layout into 8 fully
// coalesced 128B non-temporal global stores per tile. Pure bandwidth kernel:
// ~0.25 GiB read + 1 GiB NT write -> ~57us at 23.3 TB/s.

typedef __attribute__((ext_vector_type(2))) float v2f;
typedef __attribute__((ext_vector_type(8))) float v8f;

#define B_  4
#define C_  64            // wavelet groups
#define H_  256
#define W_  256
#define CT_ (4 * C_)      // 256 input channels
#define OH_ (2 * H_)
#define OW_ (2 * W_)
#define ROWSTRIDE 33      // padded LDS row stride (dwords): conflict-free both phases
#define WAVE_LDS  (8 * ROWSTRIDE)

__global__ __launch_bounds__(256) void idwt_haar_wmma(
    const float* __restrict__ x,    // [B, 4C, H, W]
    const float* __restrict__ fil,  // [4,2,2] flat; index j*4+a*2+c
    float* __restrict__ out)        // [B, 4C, 2H, 2W]
{
    __shared__ float lds[8 * WAVE_LDS];

    const int lane = threadIdx.x & 31;
    const int wib  = threadIdx.x >> 5;           // wave within block
    const int wave = blockIdx.x * 8 + wib;       // global wave id

    // wave -> (b, g, h); one wave handles a full W row of one group plane
    const int h = wave & (H_ - 1);
    const int t = wave >> 8;
    const int g = t & (C_ - 1);
    const int b = t >> 6;

    // ---- B matrix: B[k][n] = fil[n], uniform over K -> both VGPRs = fil[lane&15]
    const int   q  = lane & 15;                  // n = j*4 + a*2 + c
    const float fv = fil[q];
    v2f bm; bm.x = fv; bm.y = fv;

    // ---- A matrix pointers (ISA 16x4 f32 layout):
    // lanes 0-15: v0->K=0, v1->K=1 ; lanes 16-31: v0->K=2, v1->K=3 ; M = lane&15
    const int m  = lane & 15;                    // pixel within 16-wide tile
    const int kb = (lane >> 4) << 1;             // K base: 0 or 2
    const size_t ch0 = (size_t)(b * CT_ + kb * C_ + g);
    const float* p0 = x + (ch0 * H_ + h) * (size_t)W_ + m;
    const float* p1 = p0 + (size_t)C_ * H_ * W_; // channel n -> n+1 is +C_ planes

    // D-tile scatter coordinates: VGPR r holds M = r + 8*(lane>>4), N = lane&15
    float* lw = lds + wib * WAVE_LDS;
    const int row = q >> 1;                      // (j*2 + a)
    const int cc  = q & 1;                       // c
    const int Mb  = (lane >> 4) << 3;            // 0 or 8

    const size_t obase_ch = (size_t)(b * CT_ + g * 4);   // + j
    const v8f cz = {};

    for (int w0 = 0; w0 < W_; w0 += 16) {
        v2f a;
        a.x = __builtin_nontemporal_load(p0 + w0);
        a.y = __builtin_nontemporal_load(p1 + w0);

        // D[M,N] = sum_k A[M,k] * fil[N]  ==  s[pixel] * filter[n]
        v8f d = __builtin_amdgcn_wmma_f32_16x16x4_f32(
            false, a, false, bm, (short)0, cz, false, false);

        // Scatter the 16x16 tile into LDS as [row=(j,a)][2*M + c]
#pragma unroll
        for (int r = 0; r < 8; ++r) {
            lw[row * ROWSTRIDE + ((Mb + r) << 1) + cc] = d[r];
        }
        // same-wave LDS is in-order: no barrier needed before read-back

        // 8 coalesced 128B NT stores: one 32-float output row segment per pass
#pragma unroll
        for (int rr = 0; rr < 8; ++rr) {
            float v = lw[rr * ROWSTRIDE + lane];
            const int j  = rr >> 1;
            const int aa = rr & 1;
            size_t o = ((obase_ch + j) * OH_ + (size_t)(2 * h + aa)) * OW_
                     + (size_t)(2 * w0 + lane);
            __builtin_nontemporal_store(v, out + o);
        }
    }
}

extern "C" void kernel_launch(void* const* d_in, const int* in_sizes, int n_in,
                              void* d_out, int out_size, void* d_ws, size_t ws_size,
                              hipStream_t stream) {
    const float* x   = (const float*)d_in[0];
    const float* fil = (const float*)d_in[1];
    float*       out = (float*)d_out;

    // one wave per (b, g, h): 4*64*256 = 65536 waves; 8 waves / 256-thread block
    const int total_waves = B_ * C_ * H_;
    dim3 grid(total_waves / 8), block(256);
    idwt_haar_wmma<<<grid, block, 0, stream>>>(x, fil, out);

    (void)in_sizes; (void)n_in; (void)out_size; (void)d_ws; (void)ws_size;
}